// SmolGCN_23304492548303
// MI455X (gfx1250) — compile-verified
//
#include <hip/hip_runtime.h>
#include <hip/hip_bf16.h>
#include <math.h>

typedef __attribute__((ext_vector_type(2))) float v2f;
typedef __attribute__((ext_vector_type(8))) float v8f;

// ---------------------------------------------------------------------------
// Small elementwise / setup kernels
// ---------------------------------------------------------------------------

__global__ void k_fill(float* __restrict__ p, float v, int n) {
  int i = blockIdx.x * blockDim.x + threadIdx.x;
  if (i < n) p[i] = v;
}

__global__ void k_deg_accum(const long long* __restrict__ dst,
                            const float* __restrict__ w,
                            float* __restrict__ deg, int E) {
  int e = blockIdx.x * blockDim.x + threadIdx.x;
  if (e < E) atomicAdd(&deg[dst[e]], w[e]);
}

__global__ void k_rsqrt(float* __restrict__ deg, int n) {
  int i = blockIdx.x * blockDim.x + threadIdx.x;
  if (i < n) {
    float d = deg[i];
    deg[i] = (d > 0.f) ? rsqrtf(d) : 0.f;
  }
}

// Bp[k*32+c] = (k<bk && c<bn) ? B[k*ldb+c] : 0   -- zero-padded weight table
__global__ void k_padB(const float* __restrict__ B, int ldb, int bk, int bn,
                       float* __restrict__ Bp, int K) {
  int i = blockIdx.x * blockDim.x + threadIdx.x;
  if (i < K * 32) {
    int k = i >> 5, c = i & 31;
    Bp[i] = (k < bk && c < bn) ? B[(size_t)k * ldb + c] : 0.f;
  }
}

// out[n*32+c] = dinv[n]^2 * feat[n*32+c]   (self-loop contribution)
__global__ void k_self_init(const float* __restrict__ feat,
                            const float* __restrict__ dinv,
                            float* __restrict__ outp, int N) {
  int i = blockIdx.x * blockDim.x + threadIdx.x;
  if (i < N * 32) {
    int n = i >> 5;
    float di = dinv[n];
    outp[i] = di * di * feat[i];
  }
}

// buf[n*32+c] = c < F ? relu(buf + b[c]) : 0   (keep padding columns zero)
__global__ void k_bias_relu_pad(float* __restrict__ buf, const float* __restrict__ b,
                                int N, int F) {
  int i = blockIdx.x * blockDim.x + threadIdx.x;
  if (i < N * 32) {
    int c = i & 31;
    buf[i] = (c < F) ? fmaxf(buf[i] + b[c], 0.f) : 0.f;
  }
}

// One wave-lane per (edge, feature). acc[dst*32+c] += dinv[s]*w*dinv[d]*feat[src*32+c]
__global__ __launch_bounds__(256) void k_scatter(
    const long long* __restrict__ src, const long long* __restrict__ dst,
    const float* __restrict__ w, const float* __restrict__ dinv,
    const float* __restrict__ feat, float* __restrict__ acc,
    int E, int F) {
  long long t = (long long)blockIdx.x * blockDim.x + threadIdx.x;
  long long e = t >> 5;
  int c = (int)(t & 31);
  if (e < E && c < F) {
    long long s = src[e], d = dst[e];
    float coef = dinv[s] * w[e] * dinv[d];
    atomicAdd(&acc[d * 32 + c], coef * feat[s * 32 + c]);
  }
}

// ---------------------------------------------------------------------------
// FP32 WMMA GEMM:  C[M x 32] = A[M x K] * Bp[K x 32]  (Bp pre-padded, no
// guards in the hot loop; A rows clamped so loads are unconditional).
// Block = 4 waves; wave w: ntile = w&1, two M-tiles (B fragment reused).
// Fully unrolled K loop -> batched loads, back-to-back v_wmma_f32_16x16x4_f32.
// ---------------------------------------------------------------------------
template <int K, int LDA>
__global__ __launch_bounds__(128) void k_gemm_wmma(
    const float* __restrict__ A, int M,
    const float* __restrict__ Bp,
    float* __restrict__ C) {
  const int wave  = threadIdx.x >> 5;
  const int lane  = threadIdx.x & 31;
  const int ntile = wave & 1;
  const int mt0   = blockIdx.x * 4 + (wave >> 1) * 2;   // first of 2 m-tiles
  const int row0  = mt0 * 16;
  if (row0 >= M) return;

  const int mlane = lane & 15;
  const int khi   = lane >> 4;              // 0: lanes 0-15, 1: lanes 16-31
  const int bcol  = ntile * 16 + mlane;

  int arow0 = row0 + mlane;      if (arow0 >= M) arow0 = M - 1;   // clamp (stores guarded)
  int arow1 = row0 + 16 + mlane; if (arow1 >= M) arow1 = M - 1;

  const float* a0p = A + (size_t)arow0 * LDA + khi * 2;
  const float* a1p = A + (size_t)arow1 * LDA + khi * 2;
  const float* bp  = Bp + (khi * 2) * 32 + bcol;

  v8f acc0 = {};
  v8f acc1 = {};
#pragma unroll
  for (int k = 0; k < K; k += 4) {
    const float2 t0 = *reinterpret_cast<const float2*>(a0p + k);
    const float2 t1 = *reinterpret_cast<const float2*>(a1p + k);
    v2f a0; a0.x = t0.x; a0.y = t0.y;
    v2f a1; a1.x = t1.x; a1.y = t1.y;
    v2f b;  b.x = bp[k * 32]; b.y = bp[k * 32 + 32];
    acc0 = __builtin_amdgcn_wmma_f32_16x16x4_f32(false, a0, false, b,
                                                 (short)0, acc0, false, false);
    acc1 = __builtin_amdgcn_wmma_f32_16x16x4_f32(false, a1, false, b,
                                                 (short)0, acc1, false, false);
  }

#pragma unroll
  for (int r = 0; r < 8; ++r) {
    int s0 = row0 + r + 8 * khi;
    int s1 = s0 + 16;
    if (s0 < M) C[(size_t)s0 * 32 + bcol] = acc0[r];
    if (s1 < M) C[(size_t)s1 * 32 + bcol] = acc1[r];
  }
}

// ---------------------------------------------------------------------------
// bias + relu + log_softmax, one wave32 per node (lane = class)
// ---------------------------------------------------------------------------
__global__ __launch_bounds__(256) void k_logsoftmax(float* __restrict__ out,
                                                    const float* __restrict__ b2,
                                                    int N, int C) {
  int wid  = blockIdx.x * (blockDim.x >> 5) + (threadIdx.x >> 5);
  int lane = threadIdx.x & 31;
  if (wid >= N) return;
  bool ok = lane < C;
  float v = ok ? fmaxf(out[(size_t)wid * 32 + lane] + b2[lane], 0.f) : 0.f;
  float m = ok ? v : -3.0e38f;
  for (int off = 16; off; off >>= 1) m = fmaxf(m, __shfl_xor(m, off, 32));
  float e = ok ? expf(v - m) : 0.f;
  float s = e;
  for (int off = 16; off; off >>= 1) s += __shfl_xor(s, off, 32);
  if (ok) out[(size_t)wid * 32 + lane] = v - m - logf(s);
}

// ---------------------------------------------------------------------------
// Launch
// ---------------------------------------------------------------------------
extern "C" void kernel_launch(void* const* d_in, const int* in_sizes, int n_in,
                              void* d_out, int out_size, void* d_ws, size_t ws_size,
                              hipStream_t stream) {
  const float*     x   = (const float*)d_in[0];
  const long long* ei  = (const long long*)d_in[1];   // int64 [2, E]
  const float*     ew  = (const float*)d_in[2];
  const float*     W1  = (const float*)d_in[3];       // [nfeat, nhid]
  const float*     b1  = (const float*)d_in[4];
  const float*     W2  = (const float*)d_in[5];       // [nhid, ncls]
  const float*     b2  = (const float*)d_in[6];

  const int nhid  = in_sizes[4];            // 20
  const int ncls  = in_sizes[6];            // 32
  const int nfeat = in_sizes[3] / nhid;     // 128
  const int N     = in_sizes[0] / nfeat;    // 100000
  const int E     = in_sizes[2];            // 3200000
  const long long* esrc = ei;
  const long long* edst = ei + E;

  // workspace: dinv[N] | hW[N*32] | x1[N*32] | Bp1[128*32] | Bp2[32*32]
  float* dinv = (float*)d_ws;
  float* hW   = dinv + N;
  float* x1   = hW + (size_t)N * 32;
  float* Bp1  = x1 + (size_t)N * 32;
  float* Bp2  = Bp1 + 128 * 32;
  float* out  = (float*)d_out;              // N*32, scatter target for layer 2

  const int T = 256;
  const int nb_node = (N + T - 1) / T;
  const int nb_feat = (N * 32 + T - 1) / T;
  const int nb_edge = (E + T - 1) / T;
  const long long tot = (long long)E * 32;
  const unsigned nb_scat = (unsigned)((tot + T - 1) / T);
  const int nb_gemm = (N + 63) / 64;        // 4 m-tiles per block

  // degrees & symmetric normalization
  k_fill<<<nb_node, T, 0, stream>>>(dinv, 1.0f, N);                 // self loop
  k_deg_accum<<<nb_edge, T, 0, stream>>>(edst, ew, dinv, E);
  k_rsqrt<<<nb_node, T, 0, stream>>>(dinv, N);

  // zero-padded weight tables (W1 -> 128x32, W2 -> 32x32)
  k_padB<<<(128 * 32 + T - 1) / T, T, 0, stream>>>(W1, nhid, nfeat, nhid, Bp1, 128);
  k_padB<<<(32 * 32 + T - 1) / T, T, 0, stream>>>(W2, ncls, nhid, ncls, Bp2, 32);

  // layer 1: hW = x @ W1, aggregate, bias+relu
  k_gemm_wmma<128, 128><<<nb_gemm, 128, 0, stream>>>(x, N, Bp1, hW);
  k_self_init<<<nb_feat, T, 0, stream>>>(hW, dinv, x1, N);
  k_scatter<<<nb_scat, T, 0, stream>>>(esrc, edst, ew, dinv, hW, x1, E, nhid);
  k_bias_relu_pad<<<nb_feat, T, 0, stream>>>(x1, b1, N, nhid);

  // layer 2: hW = x1 @ W2 (K padded to 32), aggregate into out
  k_gemm_wmma<32, 32><<<nb_gemm, 128, 0, stream>>>(x1, N, Bp2, hW);
  k_self_init<<<nb_feat, T, 0, stream>>>(hW, dinv, out, N);
  k_scatter<<<nb_scat, T, 0, stream>>>(esrc, edst, ew, dinv, hW, out, E, ncls);

  // bias + relu + log_softmax in place
  k_logsoftmax<<<(N + 7) / 8, T, 0, stream>>>(out, b2, N, ncls);

  (void)n_in; (void)out_size; (void)ws_size;
}